// AssociationLayer_64372969832990
// MI455X (gfx1250) — compile-verified
//
#include <hip/hip_runtime.h>

// Sinkhorn association layer for MI455X (gfx1250, wave32).
// One workgroup per batch example; K (=exp(lambda*aff), ~300KB f32) is kept
// resident in the 320KB WGP LDS across all 100 iterations. Matvecs run on
// V_WMMA_F32_16X16X4_F32 with the v-chunk broadcast across all 16 B columns.

#define TPB 256

constexpr int MT    = 256;   // padded tracklets/detections in reference
constexpr int RP    = 272;   // padded rows (17 tiles of 16, covers 257)
constexpr int CP    = 272;   // padded cols
constexpr int PITCH = 276;   // LDS row stride in floats; 276%64=20 -> 16-row
                             // strided reads hit 16 distinct banks
constexpr int NIT   = 100;
constexpr float LAMB = 10.0f;

typedef float v2f __attribute__((ext_vector_type(2)));
typedef float v8f __attribute__((ext_vector_type(8)));

// transport value with fixed rounding order so the max pass and the output
// pass produce bit-identical values (assignment uses float equality).
__device__ __forceinline__ float tval(float ui, float k, float vj) {
  return __fmul_rn(__fmul_rn(ui, k), vj);
}

__global__ __launch_bounds__(TPB) void sinkhorn_assoc_kernel(
    const float* __restrict__ aff,   // [B,256,256]
    const int* __restrict__ ndet,    // [B]
    const int* __restrict__ ntrk,    // [B]
    float* __restrict__ outS,        // [B,65536] transport
    float* __restrict__ outA) {      // [B,65536] assignment (0/1)
  extern __shared__ float smem[];
  float* Ks   = smem;                 // RP*PITCH
  float* u_s  = Ks + RP * PITCH;      // RP
  float* v_s  = u_s + RP;             // RP
  float* y_s  = v_s + RP;             // RP (matvec result buffer)
  float* rt_s = y_s + RP;             // RP (row targets)
  float* ct_s = rt_s + RP;            // RP (col targets)
  float* rm_s = ct_s + RP;            // RP (row max)
  float* cm_s = rm_s + RP;            // RP (col max)

  const int b    = blockIdx.x;
  const int tid  = threadIdx.x;
  const int lane = tid & 31;
  const int wave = tid >> 5;
  const int nt   = ntrk[b];
  const int nd   = ndet[b];

  // ---- zero K (incl. padding); init u=v=1 and row/col targets ----
  for (int x = tid; x < RP * PITCH; x += TPB) Ks[x] = 0.f;
  for (int i = tid; i < RP; i += TPB) {
    u_s[i] = 1.f;
    v_s[i] = 1.f;
    rt_s[i] = (i < nt) ? 1.f : ((i == nt) ? (float)nt : 0.f);
    ct_s[i] = (i < nd) ? 1.f : ((i == nd) ? (float)nd : 0.f);
  }
  __syncthreads();

  // ---- K = exp(lambda*aff) inside [nt,nd); 1.0 on birth/death row+col ----
  const float* affb = aff + ((size_t)b << 16);
  for (int i = 0; i <= nt; ++i) {
    for (int j = tid; j <= nd; j += TPB) {
      float kv = 1.f;
      if (i < nt && j < nd) kv = __expf(LAMB * affb[(i << 8) + j]);
      Ks[i * PITCH + j] = kv;
    }
  }
  __syncthreads();

  // WMMA 16x16x4 f32 A-layout: lanes 0-15 hold rows m=0..15 with k={0,1},
  // lanes 16-31 the same rows with k={2,3}. B is the v-chunk broadcast
  // across all 16 columns, so every column of D carries the same partial y.
  const int koff = (lane >> 4) << 1;   // 0 or 2
  const int mrow = lane & 15;

  for (int it = 0; it < NIT; ++it) {
    // ---- y = K @ v (row tiles striped across the 8 waves) ----
    for (int t = wave; t < RP / 16; t += TPB / 32) {
      const float* arow = Ks + (t * 16 + mrow) * PITCH;
      v8f acc = {};
      for (int kk = 0; kk < CP; kk += 4) {
        v2f a;  a.x  = arow[kk + koff];  a.y  = arow[kk + koff + 1];
        v2f bb; bb.x = v_s[kk + koff];   bb.y = v_s[kk + koff + 1];
        acc = __builtin_amdgcn_wmma_f32_16x16x4_f32(
            false, a, false, bb, (short)0, acc, false, false);
      }
      if (mrow == 0) {  // D: vgpr p -> row p (lanes<16) / p+8 (lanes>=16)
        int base = t * 16 + (lane >> 4) * 8;
        #pragma unroll
        for (int p = 0; p < 8; ++p) y_s[base + p] = acc[p];
      }
    }
    __syncthreads();
    for (int i = tid; i < RP; i += TPB) {        // u = safe_div(rt, y)
      float d = y_s[i];
      u_s[i] = (d > 0.f) ? rt_s[i] / d : 0.f;
    }
    __syncthreads();

    // ---- z = K^T @ u (col tiles; A loads read K transposed) ----
    for (int t = wave; t < CP / 16; t += TPB / 32) {
      const int col = t * 16 + mrow;
      v8f acc = {};
      for (int kk = 0; kk < RP; kk += 4) {
        v2f a;
        a.x = Ks[(kk + koff) * PITCH + col];
        a.y = Ks[(kk + koff + 1) * PITCH + col];
        v2f bb; bb.x = u_s[kk + koff]; bb.y = u_s[kk + koff + 1];
        acc = __builtin_amdgcn_wmma_f32_16x16x4_f32(
            false, a, false, bb, (short)0, acc, false, false);
      }
      if (mrow == 0) {
        int base = t * 16 + (lane >> 4) * 8;
        #pragma unroll
        for (int p = 0; p < 8; ++p) y_s[base + p] = acc[p];
      }
    }
    __syncthreads();
    for (int j = tid; j < CP; j += TPB) {        // v = safe_div(ct, z)
      float d = y_s[j];
      v_s[j] = (d > 0.f) ? ct_s[j] / d : 0.f;
    }
    __syncthreads();
  }

  // ---- row/col maxima of transport = u_i * K_ij * v_j (>=0 everywhere) ----
  for (int i = tid; i < RP; i += TPB) {
    float ui = u_s[i], m = 0.f;
    const float* arow = Ks + i * PITCH;
    for (int j = 0; j < CP; ++j) m = fmaxf(m, tval(ui, arow[j], v_s[j]));
    rm_s[i] = m;
  }
  for (int j = tid; j < CP; j += TPB) {
    float vj = v_s[j], m = 0.f;
    for (int i = 0; i < RP; ++i) m = fmaxf(m, tval(u_s[i], Ks[i * PITCH + j], vj));
    cm_s[j] = m;
  }
  __syncthreads();

  // ---- pack [nt,nd) block row-major, zero-pad to MT*MT ----
  const size_t obase = (size_t)b << 16;
  for (int i = 0; i < nt; ++i) {
    float ui = u_s[i], rm = rm_s[i];
    const float* arow = Ks + i * PITCH;
    size_t rowbase = obase + (size_t)i * nd;
    for (int j = tid; j < nd; j += TPB) {
      float t = tval(ui, arow[j], v_s[j]);
      bool asn = (t == rm) && (t == cm_s[j]) && (t > 0.f);
      outS[rowbase + j] = t;
      outA[rowbase + j] = asn ? 1.f : 0.f;
    }
  }
  for (int x = nt * nd + tid; x < MT * MT; x += TPB) {
    outS[obase + x] = 0.f;
    outA[obase + x] = 0.f;
  }
}

extern "C" void kernel_launch(void* const* d_in, const int* in_sizes, int n_in,
                              void* d_out, int out_size, void* d_ws, size_t ws_size,
                              hipStream_t stream) {
  const float* aff  = (const float*)d_in[0];  // affinity_scores [B,256,256]
  const int*   ndet = (const int*)d_in[1];    // num_detections  [B]
  const int*   ntrk = (const int*)d_in[2];    // num_tracklets   [B]
  const int B = in_sizes[1];

  float* outS = (float*)d_out;                     // sinkhorn_dense
  float* outA = outS + (size_t)B * MT * MT;        // assignment_dense (0/1)

  const size_t ldsBytes = (size_t)(RP * PITCH + 7 * RP) * sizeof(float); // ~301KB
  sinkhorn_assoc_kernel<<<B, TPB, ldsBytes, stream>>>(aff, ndet, ntrk, outS, outA);
}